// SE3GroupConvLayer_16767552323912
// MI455X (gfx1250) — compile-verified
//
#include <hip/hip_runtime.h>
#include <math.h>

#define N_CH   16
#define FEAT   64
#define HID    64
#define NBOT   16
#define MPOSE  4         // M is fixed by the reference setup (B=8, M=4)
#define EPSL   1e-8f

typedef float v2f __attribute__((ext_vector_type(2)));
typedef float v8f __attribute__((ext_vector_type(8)));

// fast silu: v_exp_f32 + v_rcp_f32 (~2ulp) instead of libm expf + sw-divide
__device__ __forceinline__ float silu_f(float x) {
  return x * __builtin_amdgcn_rcpf(1.0f + __expf(-x));
}

// async copy of one 16B chunk global -> LDS (ASYNCcnt-tracked, no VGPR round trip)
__device__ __forceinline__ void async_copy_b128(void* lds_dst, const void* gsrc) {
  unsigned lds_off = (unsigned)(unsigned long long)lds_dst;   // flat addr low 32 = LDS offset
  unsigned long long ga = (unsigned long long)gsrc;
  asm volatile("global_load_async_to_lds_b128 %0, %1, off"
               :: "v"(lds_off), "v"(ga) : "memory");
}
__device__ __forceinline__ void async_wait0() {
  asm volatile("s_wait_asynccnt 0" ::: "memory");
}

// ---------------------------------------------------------------- K1: poses -> D1_inv / D1_g
__global__ void k_poses(const float* __restrict__ poses, float* __restrict__ Dinv,
                        float* __restrict__ Dg, int BM) {
  int t = blockIdx.x * blockDim.x + threadIdx.x;
  if (t >= BM) return;
  float a[6];
#pragma unroll
  for (int i = 0; i < 6; ++i) {
    float x = poses[t * 6 + i];
    a[i] = fminf(10.0f, fmaxf(-10.0f, x));
  }
  float wx = a[3], wy = a[4], wz = a[5];
  float t2 = wx * wx + wy * wy + wz * wz + 1e-12f;
  float th = sqrtf(t2);
  float sA = sinf(th) / th;
  float cB = (1.0f - cosf(th)) / t2;
  float K[9] = {0.f, -wz, wy, wz, 0.f, -wx, -wy, wx, 0.f};
  float K2[9];
#pragma unroll
  for (int r = 0; r < 3; ++r)
#pragma unroll
    for (int c = 0; c < 3; ++c) {
      float s = 0.f;
#pragma unroll
      for (int k = 0; k < 3; ++k) s += K[r * 3 + k] * K[k * 3 + c];
      K2[r * 3 + c] = s;
    }
  float R[9];
#pragma unroll
  for (int r = 0; r < 3; ++r)
#pragma unroll
    for (int c = 0; c < 3; ++c)
      R[r * 3 + c] = ((r == c) ? 1.0f : 0.0f) + sA * K[r * 3 + c] + cB * K2[r * 3 + c];
  const int p[3] = {1, 2, 0};
#pragma unroll
  for (int i = 0; i < 3; ++i)
#pragma unroll
    for (int j = 0; j < 3; ++j) {
      Dg[t * 9 + i * 3 + j]   = R[p[i] * 3 + p[j]];        // D1 of R
      Dinv[t * 9 + i * 3 + j] = R[p[j] * 3 + p[i]];        // D1 of R^T
    }
}

// ---------------------------------------------------------------- K2: rotate features into `cur`
__global__ void k_rotate(const float* __restrict__ feat, const int* __restrict__ bidx,
                         const float* __restrict__ Dinv, float* __restrict__ cur,
                         int N, int Nsb) {
  int idx = blockIdx.x * blockDim.x + threadIdx.x;
  if (idx >= Nsb * N_CH) return;
  int c = idx & 15;
  int nsb = idx >> 4;
  int m = nsb / N;
  int n = nsb - m * N;
  const float* f = feat + (size_t)n * FEAT;
  float s  = f[c];
  float v0 = f[16 + c * 3 + 0];
  float v1 = f[16 + c * 3 + 1];
  float v2 = f[16 + c * 3 + 2];
  int p = bidx[n] * MPOSE + m;
  const float* D = Dinv + p * 9;
  float r0 = D[0] * v0 + D[1] * v1 + D[2] * v2;
  float r1 = D[3] * v0 + D[4] * v1 + D[5] * v2;
  float r2 = D[6] * v0 + D[7] * v1 + D[8] * v2;
  if (r0 != r0) r0 = v0;   // NaN guard (reference: where(isnan(aligned), feat, aligned))
  if (r1 != r1) r1 = v1;
  if (r2 != r2) r2 = v2;
  float* o = cur + (size_t)nsb * FEAT;
  o[c] = s;
  o[16 + c * 3 + 0] = r0;
  o[16 + c * 3 + 1] = r1;
  o[16 + c * 3 + 2] = r2;
}

// ---------------------------------------------------------------- K3: per-edge geometry + radial MLP (WMMA fp32)
#define H1S 65   // padded LDS stride for the h1 transpose buffer
__global__ void __launch_bounds__(256)
k_edge_mlp(const float* __restrict__ pos, const int* __restrict__ eidx,
           const float* __restrict__ rw0, const float* __restrict__ rb0,
           const float* __restrict__ rw1, const float* __restrict__ rb1,
           const float* __restrict__ rw2, const float* __restrict__ rb2,
           float* __restrict__ sh_out, float* __restrict__ remb_out, int E) {
  __shared__ __align__(16) float rw1s[HID * HID];
  __shared__ __align__(16) float rw2s[HID * NBOT];
  __shared__ float rb1s[HID];
  __shared__ float rb2s[NBOT];
  __shared__ float h1s[8][16 * H1S];

  // async-stage the weight matrices into LDS (16KB + 4KB), b128 per transfer
  for (int i = threadIdx.x; i < (HID * HID) / 4; i += blockDim.x)
    async_copy_b128(&rw1s[i * 4], rw1 + i * 4);
  for (int i = threadIdx.x; i < (HID * NBOT) / 4; i += blockDim.x)
    async_copy_b128(&rw2s[i * 4], rw2 + i * 4);
  if (threadIdx.x < HID)  rb1s[threadIdx.x] = rb1[threadIdx.x];
  if (threadIdx.x < NBOT) rb2s[threadIdx.x] = rb2[threadIdx.x];
  async_wait0();
  __syncthreads();

  const int wave = threadIdx.x >> 5;
  const int lane = threadIdx.x & 31;
  const int r    = lane & 15;       // edge row within tile (also N-col index)
  const int hi   = lane >> 4;       // half-wave selector
  const int tile = blockIdx.x * 8 + wave;
  const int e_base = tile * 16;
  const int e_real = e_base + r;
  const int e = (e_real < E) ? e_real : (E - 1);

  // ---- geometry (both half-waves compute the same 16 edges; lanes 16-31 duplicate)
  int row = eidx[e], col = eidx[E + e];
  float ex = pos[row * 3 + 0] - pos[col * 3 + 0];
  float ey = pos[row * 3 + 1] - pos[col * 3 + 1];
  float ez = pos[row * 3 + 2] - pos[col * 3 + 2];
  float el = sqrtf(ex * ex + ey * ey + ez * ez);
  bool  valid = el > EPSL;
  float inv = valid ? (1.0f / el) : 1.0f;
  const float SQ3 = 1.7320508075688772f;
  float Y1x = SQ3 * (ey * inv);   // unit[[1,2,0]] permutation
  float Y1y = SQ3 * (ez * inv);
  float Y1z = SQ3 * (ex * inv);
  if (hi == 0 && e_real < E) {
    sh_out[e * 4 + 0] = valid ? 1.0f : 0.0f;
    sh_out[e * 4 + 1] = valid ? Y1x : 0.0f;
    sh_out[e * 4 + 2] = valid ? Y1y : 0.0f;
    sh_out[e * 4 + 3] = valid ? Y1z : 0.0f;
  }

  // ---- GEMM1: h1_pre = h0(16x64) @ rw1(64x64);  h0 generated on the fly from el
  v8f z = {0.f, 0.f, 0.f, 0.f, 0.f, 0.f, 0.f, 0.f};
  v8f acc[4] = {z, z, z, z};
  for (int k0 = 0; k0 < HID; k0 += 4) {
    int kA = k0 + 2 * hi;
    v2f a;
    a.x = silu_f(el * rw0[kA] + rb0[kA]);
    a.y = silu_f(el * rw0[kA + 1] + rb0[kA + 1]);
#pragma unroll
    for (int nb = 0; nb < 4; ++nb) {
      int n = nb * 16 + r;
      v2f b;
      b.x = rw1s[kA * HID + n];
      b.y = rw1s[(kA + 1) * HID + n];
      acc[nb] = __builtin_amdgcn_wmma_f32_16x16x4_f32(
          false, a, false, b, (short)0, acc[nb], false, false);
    }
  }

  // ---- h1 = silu(h1_pre + rb1) -> LDS (transpose from C-layout to A-layout)
  float* hw = h1s[wave];
#pragma unroll
  for (int nb = 0; nb < 4; ++nb) {
    int n = nb * 16 + r;
    float bv = rb1s[n];
#pragma unroll
    for (int j = 0; j < 8; ++j) {
      int rowi = j + 8 * hi;
      hw[rowi * H1S + n] = silu_f(acc[nb][j] + bv);
    }
  }
  __syncthreads();

  // ---- GEMM2: remb_pre = h1(16x64) @ rw2(64x16)
  v8f acc2 = z;
  for (int k0 = 0; k0 < HID; k0 += 4) {
    int kA = k0 + 2 * hi;
    v2f a;
    a.x = hw[r * H1S + kA];
    a.y = hw[r * H1S + kA + 1];
    v2f b;
    b.x = rw2s[kA * NBOT + r];
    b.y = rw2s[(kA + 1) * NBOT + r];
    acc2 = __builtin_amdgcn_wmma_f32_16x16x4_f32(
        false, a, false, b, (short)0, acc2, false, false);
  }
  float bv2 = rb2s[r];
#pragma unroll
  for (int j = 0; j < 8; ++j) {
    int rowi = j + 8 * hi;
    int eo = e_base + rowi;
    if (eo < E) remb_out[(size_t)eo * NBOT + r] = acc2[j] + bv2;
  }
}

// ---------------------------------------------------------------- K4: messages + atomic scatter-add
__global__ void __launch_bounds__(256)
k_message(const float* __restrict__ cur, const float* __restrict__ sh,
          const float* __restrict__ remb, const float* __restrict__ wp,
          const float* __restrict__ bp, const int* __restrict__ eidx,
          float* __restrict__ agg, int N, int E) {
  __shared__ __align__(16) float wps[NBOT * 64];
  __shared__ float bps[64];
  for (int i = threadIdx.x; i < (NBOT * 64) / 4; i += blockDim.x)
    async_copy_b128(&wps[i * 4], wp + i * 4);
  if (threadIdx.x < 64) bps[threadIdx.x] = bp[threadIdx.x];
  async_wait0();
  __syncthreads();

  long idx = (long)blockIdx.x * blockDim.x + threadIdx.x;
  long total = (long)MPOSE * E * 16;
  if (idx >= total) return;
  int c = (int)(idx & 15);
  long t = idx >> 4;
  int m = (int)(t / E);
  int e = (int)(t - (long)m * E);

  int row = eidx[e], col = eidx[E + e];
  size_t row_sb = (size_t)m * N + row;
  size_t col_sb = (size_t)m * N + col;

  // w = remb @ wp + bp  (only 4 output columns per thread)
  float w0 = bps[0 * 16 + c], w1 = bps[1 * 16 + c];
  float w2 = bps[2 * 16 + c], w3 = bps[3 * 16 + c];
  const float* rm = remb + (size_t)e * NBOT;
  const float* cs = cur + col_sb * FEAT;
  __builtin_prefetch(cs, 0, 3);            // global_prefetch_b8 on the gather line
#pragma unroll
  for (int d = 0; d < NBOT; ++d) {
    float rv = rm[d];
    w0 += rv * wps[d * 64 + 0 * 16 + c];
    w1 += rv * wps[d * 64 + 1 * 16 + c];
    w2 += rv * wps[d * 64 + 2 * 16 + c];
    w3 += rv * wps[d * 64 + 3 * 16 + c];
  }
  float Y0 = sh[e * 4 + 0], Yx = sh[e * 4 + 1], Yy = sh[e * 4 + 2], Yz = sh[e * 4 + 3];
  float ssrc = cs[c];
  float vx = cs[16 + c * 3 + 0];
  float vy = cs[16 + c * 3 + 1];
  float vz = cs[16 + c * 3 + 2];

  float msg_s = w0 * ssrc * Y0 + w3 * (vx * Yx + vy * Yy + vz * Yz);
  float mvx = w1 * ssrc * Yx + w2 * vx * Y0;
  float mvy = w1 * ssrc * Yy + w2 * vy * Y0;
  float mvz = w1 * ssrc * Yz + w2 * vz * Y0;

  float* ar = agg + row_sb * FEAT;
  atomicAdd(&ar[c], msg_s);
  atomicAdd(&ar[16 + c * 3 + 0], mvx);
  atomicAdd(&ar[16 + c * 3 + 1], mvy);
  atomicAdd(&ar[16 + c * 3 + 2], mvz);
}

// ---------------------------------------------------------------- K5: channel mix + pooled reduction
__global__ void k_mix_pool(const float* __restrict__ agg, const float* __restrict__ Ws,
                           const float* __restrict__ Wv, const int* __restrict__ bidx,
                           float* __restrict__ pooled, float* __restrict__ cnt,
                           int N, int Nsb) {
  __shared__ float Wss[256], Wvs[256];
  if (threadIdx.x < 256) { Wss[threadIdx.x] = Ws[threadIdx.x]; Wvs[threadIdx.x] = Wv[threadIdx.x]; }
  __syncthreads();
  long idx = (long)blockIdx.x * blockDim.x + threadIdx.x;
  if (idx >= (long)Nsb * 16) return;
  int c = (int)(idx & 15);
  int nsb = (int)(idx >> 4);
  int m = nsb / N;
  int n = nsb - m * N;
  const float* ar = agg + (size_t)nsb * FEAT;
  float os = 0.f, ov0 = 0.f, ov1 = 0.f, ov2 = 0.f;
#pragma unroll
  for (int d = 0; d < N_CH; ++d) {
    os += ar[d] * Wss[d * 16 + c];
    float wv = Wvs[d * 16 + c];
    ov0 += ar[16 + d * 3 + 0] * wv;
    ov1 += ar[16 + d * 3 + 1] * wv;
    ov2 += ar[16 + d * 3 + 2] * wv;
  }
  int p = bidx[n] * MPOSE + m;
  float* pr = pooled + (size_t)p * FEAT;
  atomicAdd(&pr[c], os);
  atomicAdd(&pr[16 + c * 3 + 0], ov0);
  atomicAdd(&pr[16 + c * 3 + 1], ov1);
  atomicAdd(&pr[16 + c * 3 + 2], ov2);
  if (c == 0) atomicAdd(&cnt[p], 1.0f);
}

// ---------------------------------------------------------------- K6: mean + final rotation -> out
__global__ void k_final(const float* __restrict__ pooled, const float* __restrict__ cnt,
                        const float* __restrict__ Dg, float* __restrict__ out, int BM) {
  int idx = blockIdx.x * blockDim.x + threadIdx.x;
  if (idx >= BM * 16) return;
  int c = idx & 15;
  int p = idx >> 4;
  float sc = 1.0f / fmaxf(cnt[p], 1.0f);
  const float* pr = pooled + (size_t)p * FEAT;
  float s  = pr[c] * sc;
  float v0 = pr[16 + c * 3 + 0] * sc;
  float v1 = pr[16 + c * 3 + 1] * sc;
  float v2 = pr[16 + c * 3 + 2] * sc;
  const float* D = Dg + p * 9;
  float r0 = D[0] * v0 + D[1] * v1 + D[2] * v2;
  float r1 = D[3] * v0 + D[4] * v1 + D[5] * v2;
  float r2 = D[6] * v0 + D[7] * v1 + D[8] * v2;
  float* o = out + (size_t)p * FEAT;
  o[c] = s;
  o[16 + c * 3 + 0] = r0;
  o[16 + c * 3 + 1] = r1;
  o[16 + c * 3 + 2] = r2;
}

// ---------------------------------------------------------------- launch
extern "C" void kernel_launch(void* const* d_in, const int* in_sizes, int n_in,
                              void* d_out, int out_size, void* d_ws, size_t ws_size,
                              hipStream_t stream) {
  const float* feat  = (const float*)d_in[0];
  const float* pos   = (const float*)d_in[1];
  const float* poses = (const float*)d_in[2];
  const int*   eidx  = (const int*)d_in[3];
  const int*   bidx  = (const int*)d_in[4];
  const float* rw0 = (const float*)d_in[5];
  const float* rb0 = (const float*)d_in[6];
  const float* rw1 = (const float*)d_in[7];
  const float* rb1 = (const float*)d_in[8];
  const float* rw2 = (const float*)d_in[9];
  const float* rb2 = (const float*)d_in[10];
  const float* wp  = (const float*)d_in[11];
  const float* bp  = (const float*)d_in[12];
  const float* Ws  = (const float*)d_in[13];
  const float* Wv  = (const float*)d_in[14];

  const int N  = in_sizes[1] / 3;
  const int E  = in_sizes[3] / 2;
  const int BM = in_sizes[2] / 6;
  const int Nsb = N * MPOSE;

  float* ws     = (float*)d_ws;
  float* Dinv   = ws;
  float* Dg     = Dinv + (size_t)BM * 9;
  float* cur    = Dg + (size_t)BM * 9;
  float* shb    = cur + (size_t)Nsb * FEAT;
  float* remb   = shb + (size_t)E * 4;
  float* agg    = remb + (size_t)E * NBOT;
  float* pooled = agg + (size_t)Nsb * FEAT;
  float* cnt    = pooled + (size_t)BM * FEAT;

  // zero the accumulators every call (graph replays reuse the workspace)
  hipMemsetAsync(agg, 0, (size_t)Nsb * FEAT * sizeof(float), stream);
  hipMemsetAsync(pooled, 0, ((size_t)BM * FEAT + BM) * sizeof(float), stream);

  k_poses<<<1, 32, 0, stream>>>(poses, Dinv, Dg, BM);
  k_rotate<<<(Nsb * 16 + 255) / 256, 256, 0, stream>>>(feat, bidx, Dinv, cur, N, Nsb);

  int tiles = (E + 15) / 16;
  k_edge_mlp<<<(tiles + 7) / 8, 256, 0, stream>>>(pos, eidx, rw0, rb0, rw1, rb1,
                                                  rw2, rb2, shb, remb, E);

  long total4 = (long)MPOSE * E * 16;
  k_message<<<(int)((total4 + 255) / 256), 256, 0, stream>>>(cur, shb, remb, wp, bp,
                                                             eidx, agg, N, E);
  k_mix_pool<<<(int)(((long)Nsb * 16 + 255) / 256), 256, 0, stream>>>(agg, Ws, Wv, bidx,
                                                                      pooled, cnt, N, Nsb);
  k_final<<<(BM * 16 + 255) / 256, 256, 0, stream>>>(pooled, cnt, Dg, (float*)d_out, BM);
}